// NMILoss_43456479101658
// MI455X (gfx1250) — compile-verified
//
#include <hip/hip_runtime.h>

// ---------------------------------------------------------------------------
// NMI (Parzen-window) loss for MI455X / gfx1250.
//   - TRANS-bound phase (164M exp evals) -> v_exp_f32 via exp2 identity.
//   - 16x256 row reductions -> V_WMMA_F32_16X16X4_F32 with ones-B matrix.
// ---------------------------------------------------------------------------

typedef __attribute__((ext_vector_type(2))) float v2f;
typedef __attribute__((ext_vector_type(8))) float v8f;

#define NPATCH 16
#define BINS   256
#define SPLIT  10           // chunks per patch in the pdf kernel
#define ELEMS  10000        // elements per patch (100x100)
#define CHUNK  (ELEMS / SPLIT)
// exp(-0.5*d^2) == exp2(KEXP * d^2),  KEXP = -0.5*log2(e)
#define KEXP   (-0.72134752044448169f)
#define INVH   (0.33333333333333333f)   // 1/bandwidth, h = 3.0
#define EPS    (1e-10f)

// ---------------------------------------------------------------------------
// Kernel 1: global min/max of y and yp (nearest-resize only repeats values,
// so min/max over the original 65536-element arrays is exact).
// ws[0..3] = {minY, maxY, minP, maxP}
// ---------------------------------------------------------------------------
__global__ void k_minmax(const float* __restrict__ y,
                         const float* __restrict__ yp,
                         float* __restrict__ ws, int n) {
    __shared__ float smnY[256], smxY[256], smnP[256], smxP[256];
    const int t = threadIdx.x;
    float mnY = 1e30f, mxY = -1e30f, mnP = 1e30f, mxP = -1e30f;
    for (int i = t; i < n; i += 256) {
        float a = y[i];  mnY = fminf(mnY, a); mxY = fmaxf(mxY, a);
        float b = yp[i]; mnP = fminf(mnP, b); mxP = fmaxf(mxP, b);
    }
    smnY[t] = mnY; smxY[t] = mxY; smnP[t] = mnP; smxP[t] = mxP;
    __syncthreads();
    for (int s = 128; s > 0; s >>= 1) {
        if (t < s) {
            smnY[t] = fminf(smnY[t], smnY[t + s]);
            smxY[t] = fmaxf(smxY[t], smxY[t + s]);
            smnP[t] = fminf(smnP[t], smnP[t + s]);
            smxP[t] = fmaxf(smxP[t], smxP[t + s]);
        }
        __syncthreads();
    }
    if (t == 0) {
        ws[0] = smnY[0]; ws[1] = smxY[0];
        ws[2] = smnP[0]; ws[3] = smxP[0];
    }
}

// ---------------------------------------------------------------------------
// Kernel 2: unnormalized Parzen pdfs. One block per (patch, chunk).
// Thread s owns bin s for three bin grids (y-grid, yp-grid, joint-grid).
// linspace runs from MAX down to MIN (faithful to the reference's swap).
// All constant scale factors (1/(2*pi*h*L)) cancel under normalization.
// part layout: [(patch*SPLIT + chunk)*3 + {0:S1,1:S2,2:Sj}][bin]
// ---------------------------------------------------------------------------
__global__ void k_pdf(const float* __restrict__ y,
                      const float* __restrict__ yp,
                      const float* __restrict__ mm,
                      float* __restrict__ part) {
    const int p = blockIdx.x;    // patch 0..15
    const int c = blockIdx.y;    // chunk 0..SPLIT-1
    const int s = threadIdx.x;   // bin   0..255

    __shared__ float tile[256];

    const float loY = mm[0], hiY = mm[1], loP = mm[2], hiP = mm[3];
    const float loJ = fminf(loY, loP), hiJ = fmaxf(hiY, hiP);
    const float fs = (float)s * (1.0f / 255.0f);
    const float xi1 = hiY + (loY - hiY) * fs;   // max -> min
    const float xi2 = hiP + (loP - hiP) * fs;
    const float xij = hiJ + (loJ - hiJ) * fs;

    float a1 = 0.0f, a2 = 0.0f, aj = 0.0f;

    const int b    = p >> 2;                 // source image
    const int goff = (p & 3) * ELEMS;        // flat offset inside 200x200 image
    const int f0   = c * CHUNK;
    const int f1   = f0 + CHUNK;
    const float* __restrict__ imgY = y  + b * 16384;
    const float* __restrict__ imgP = yp + b * 16384;

    // ---- phase 1: y elements -> own bins (a1) + joint bins (aj) ----
    for (int base = f0; base < f1; base += 256) {
        const int cnt = min(256, f1 - base);
        const int f = base + s;
        if (s < cnt) {
            const int g = goff + f;
            const int r = g / 200, cc = g - r * 200;          // resized coords
            tile[s] = imgY[((r * 16) / 25) * 128 + (cc * 16) / 25]; // 128/200=16/25
        }
        __syncthreads();
#pragma unroll 4
        for (int j = 0; j < cnt; ++j) {
            const float x = tile[j];
            const float d1 = (x - xi1) * INVH;
            const float dj = (x - xij) * INVH;
            a1 += __builtin_amdgcn_exp2f(KEXP * d1 * d1);
            aj += __builtin_amdgcn_exp2f(KEXP * dj * dj);
        }
        __syncthreads();
    }
    // ---- phase 2: yp elements -> own bins (a2) + joint bins (aj) ----
    for (int base = f0; base < f1; base += 256) {
        const int cnt = min(256, f1 - base);
        const int f = base + s;
        if (s < cnt) {
            const int g = goff + f;
            const int r = g / 200, cc = g - r * 200;
            tile[s] = imgP[((r * 16) / 25) * 128 + (cc * 16) / 25];
        }
        __syncthreads();
#pragma unroll 4
        for (int j = 0; j < cnt; ++j) {
            const float x = tile[j];
            const float d2 = (x - xi2) * INVH;
            const float dj = (x - xij) * INVH;
            a2 += __builtin_amdgcn_exp2f(KEXP * d2 * d2);
            aj += __builtin_amdgcn_exp2f(KEXP * dj * dj);
        }
        __syncthreads();
    }

    float* dst = part + ((size_t)(p * SPLIT + c) * 3) * BINS + s;
    dst[0]        = a1;
    dst[BINS]     = a2;
    dst[2 * BINS] = aj;
}

// ---------------------------------------------------------------------------
// Kernel 3: reduce partials, entropies via WMMA row-reductions, final loss.
// Row sums of a 16x256 matrix: D = A(16x4) x ones(4x16) + C accumulated over
// 64 K-chunks -> v_wmma_f32_16x16x4_f32, row sum replicated across columns.
// A-layout (ISA 7.12.2): lanes 0-15 hold K=0,1 ; lanes 16-31 hold K=2,3.
// ---------------------------------------------------------------------------
__global__ void k_final(const float* __restrict__ part, float* __restrict__ out) {
    __shared__ float S[NPATCH * BINS];
    __shared__ float T[NPATCH * BINS];
    __shared__ float rs[NPATCH];
    __shared__ float E[3][NPATCH];

    const int t = threadIdx.x;   // 0..255

    for (int a = 0; a < 3; ++a) {
        // -- deterministic sum of the SPLIT partial chunks --
        for (int idx = t; idx < NPATCH * BINS; idx += 256) {
            const int n = idx >> 8, s = idx & 255;
            float acc = 0.0f;
            for (int c = 0; c < SPLIT; ++c)
                acc += part[((size_t)(n * SPLIT + c) * 3 + a) * BINS + s];
            S[idx] = acc;
        }
        __syncthreads();

        // -- row sums via WMMA (wave 0, EXEC all ones) --
        if (t < 32) {
            const int row  = t & 15;
            const int koff = (t >> 4) << 1;   // lanes 16-31 hold K=2,3
            const int half = t >> 4;
            v2f ones; ones.x = 1.0f; ones.y = 1.0f;
            v8f acc = {};
            for (int k = 0; k < 64; ++k) {
                v2f av;
                av.x = S[row * BINS + 4 * k + koff];
                av.y = S[row * BINS + 4 * k + koff + 1];
                acc = __builtin_amdgcn_wmma_f32_16x16x4_f32(
                    false, av, false, ones, (short)0, acc, false, false);
            }
#pragma unroll
            for (int v = 0; v < 8; ++v)
                rs[v + 8 * half] = acc[v];    // replicated over cols -> benign
        }
        __syncthreads();

        // -- T = p * log2(p + EPS), p = S / rowsum --
        for (int idx = t; idx < NPATCH * BINS; idx += 256) {
            const int n = idx >> 8;
            const float pr = S[idx] / rs[n];
            T[idx] = pr * __builtin_amdgcn_logf(pr + EPS);  // v_log_f32 = log2
        }
        __syncthreads();

        // -- entropy = row sums of T via WMMA --
        if (t < 32) {
            const int row  = t & 15;
            const int koff = (t >> 4) << 1;
            const int half = t >> 4;
            v2f ones; ones.x = 1.0f; ones.y = 1.0f;
            v8f acc = {};
            for (int k = 0; k < 64; ++k) {
                v2f av;
                av.x = T[row * BINS + 4 * k + koff];
                av.y = T[row * BINS + 4 * k + koff + 1];
                acc = __builtin_amdgcn_wmma_f32_16x16x4_f32(
                    false, av, false, ones, (short)0, acc, false, false);
            }
#pragma unroll
            for (int v = 0; v < 8; ++v)
                E[a][v + 8 * half] = acc[v];
        }
        __syncthreads();
    }

    if (t == 0) {
        float loss = 0.0f;
        for (int n = 0; n < NPATCH; ++n) {
            const float e1 = E[0][n], e2 = E[1][n], ej = E[2][n];
            const float nmi = 2.0f * (e1 + e2 - ej) / (e1 + e2);
            loss += fabsf(nmi - 1.0f) * 1000.0f;
        }
        out[0] = loss * (1.0f / (float)NPATCH);
    }
}

// ---------------------------------------------------------------------------
extern "C" void kernel_launch(void* const* d_in, const int* in_sizes, int n_in,
                              void* d_out, int out_size, void* d_ws, size_t ws_size,
                              hipStream_t stream) {
    const float* y  = (const float*)d_in[0];
    const float* yp = (const float*)d_in[1];
    float* out = (float*)d_out;
    float* ws  = (float*)d_ws;

    float* mm   = ws;        // 4 floats: minY, maxY, minP, maxP
    float* part = ws + 16;   // SPLIT*16*3*256 floats (~480 KB), fully overwritten

    k_minmax<<<1, 256, 0, stream>>>(y, yp, mm, in_sizes[0]);
    k_pdf<<<dim3(NPATCH, SPLIT), 256, 0, stream>>>(y, yp, mm, part);
    k_final<<<1, 256, 0, stream>>>(part, out);
}